// RobustGNN_54150947668271
// MI455X (gfx1250) — compile-verified
//
#include <hip/hip_runtime.h>

// Problem constants (match reference)
#define N_NODES  100000
#define N_EDGES  1600000
#define F_INF    14
#define K_PAD    16
#define H_DIM    64
#define G_GRAPHS 512
#define C_OUT    2

typedef __attribute__((ext_vector_type(2))) float v2f;
typedef __attribute__((ext_vector_type(8))) float v8f;

// ---------------------------------------------------------------- utilities
__global__ void fill_kernel(float* __restrict__ p, float v, int n) {
    int t = blockIdx.x * blockDim.x + threadIdx.x;
    if (t < n) p[t] = v;
}

// deg[dst[e]] += 1  (deg pre-initialized to 1.0 for the self loop)
__global__ void deg_scatter_kernel(const int* __restrict__ dst,
                                   float* __restrict__ deg, int e) {
    int t = blockIdx.x * blockDim.x + threadIdx.x;
    if (t < e) atomicAdd(&deg[dst[t]], 1.0f);
}

// in-place: deg -> dinv_sqrt ; dinv = 1/deg
__global__ void deg_finish_kernel(float* __restrict__ deg_to_ds,
                                  float* __restrict__ dinv, int n) {
    int t = blockIdx.x * blockDim.x + threadIdx.x;
    if (t < n) {
        float d = deg_to_ds[t];
        deg_to_ds[t] = rsqrtf(d);
        dinv[t] = 1.0f / d;
    }
}

// norm[e] = ds[src] * ds[dst]
__global__ void edge_norm_kernel(const int* __restrict__ src,
                                 const int* __restrict__ dst,
                                 const float* __restrict__ ds,
                                 float* __restrict__ norm, int e) {
    int t = blockIdx.x * blockDim.x + threadIdx.x;
    if (t < e) norm[t] = ds[src[t]] * ds[dst[t]];
}

// zero-pad x[N x 14] -> xpad[N x 16]
__global__ void pad_x_kernel(const float* __restrict__ x,
                             float* __restrict__ xpad, int n) {
    int t = blockIdx.x * blockDim.x + threadIdx.x;
    if (t >= n) return;
    int i = t >> 4, f = t & 15;
    xpad[t] = (f < F_INF) ? x[i * F_INF + f] : 0.0f;
}

// zero-pad W1[14 x 64] -> W1pad[16 x 64] (row-major, same linear order)
__global__ void pad_w_kernel(const float* __restrict__ W1,
                             float* __restrict__ Wpad) {
    int t = blockIdx.x * blockDim.x + threadIdx.x;
    if (t >= K_PAD * H_DIM) return;
    int k = t >> 6;
    Wpad[t] = (k < F_INF) ? W1[t] : 0.0f;
}

// --------------------------------------- dense GEMM via WMMA + fused epilogue
// m = act(A[N x K]) @ W[K x 64]
//   Q[i,f] = m[i,f]                         (consumed by the edge scatter)
//   P[i,f] = m[i,f] * dinv[i] + bias[f]     (self-loop + bias aggregation init)
// K is compile-time (16 or 64): fully unrolled, unpredicated loop of
// V_WMMA_F32_16X16X4_F32 (full f32 precision, matches reference).
// grid.x = N/16 row tiles; block = 128 threads = 4 waves; wave w owns 16-col tile w.
template <int K, bool RELU>
__global__ void gemm_wmma_fused_kernel(const float* __restrict__ A,
                                       const float* __restrict__ W,
                                       const float* __restrict__ bias,
                                       const float* __restrict__ dinv,
                                       float* __restrict__ Q,
                                       float* __restrict__ P) {
    const int wave   = threadIdx.x >> 5;        // 0..3  -> column tile
    const int lane   = threadIdx.x & 31;
    const int lHalf  = lane >> 4;               // 0 or 1
    const int lMod   = lane & 15;
    const int rowTile = blockIdx.x << 4;        // 16 rows per block
    const int colBase = wave << 4;              // 16 cols per wave

    const int row = rowTile + lMod;
    const int col = colBase + lMod;
    const float* Arow = A + (size_t)row * K + (lHalf << 1);
    const float* Wcol = W + (size_t)(lHalf << 1) * H_DIM + col;

    v8f acc = {};
#pragma unroll
    for (int k0 = 0; k0 < K; k0 += 4) {
        // A fragment (16x4): lanes 0-15 hold K=k0+{0,1}; lanes 16-31 hold K=k0+{2,3}
        float2 a = *(const float2*)(Arow + k0);
        if (RELU) { a.x = fmaxf(a.x, 0.0f); a.y = fmaxf(a.y, 0.0f); }
        // B fragment (4x16): lane holds W[k0+2*lHalf + {0,1}][col]
        float b0 = Wcol[(size_t)k0 * H_DIM];
        float b1 = Wcol[(size_t)(k0 + 1) * H_DIM];
        v2f av; av.x = a.x; av.y = a.y;
        v2f bv; bv.x = b0;  bv.y = b1;
        acc = __builtin_amdgcn_wmma_f32_16x16x4_f32(
            /*neg_a=*/false, av, /*neg_b=*/false, bv,
            /*c_mod=*/(short)0, acc, /*reuse_a=*/false, /*reuse_b=*/false);
    }
    // C/D layout: VGPR r, lanes 0-15 -> M=r, N=lane; lanes 16-31 -> M=8+r
    const float bcol  = bias[col];
    const int   row0  = rowTile + lHalf * 8;
    const size_t base = (size_t)row0 * H_DIM + col;
#pragma unroll
    for (int r = 0; r < 8; ++r) {
        float m  = acc[r];
        float di = dinv[row0 + r];      // uniform across the 16-lane half -> broadcast
        Q[base + (size_t)r * H_DIM] = m;
        P[base + (size_t)r * H_DIM] = m * di + bcol;
    }
}

// -------------------------------------------------------------- edge scatter
// 16 threads per edge, float4 gather of Q[src] (L2 resident), f32 atomic adds.
__global__ void edge_scatter_kernel(const float* __restrict__ M,
                                    const int* __restrict__ src,
                                    const int* __restrict__ dst,
                                    const float* __restrict__ norm,
                                    float* __restrict__ Agg, int nthreads) {
    int t = blockIdx.x * blockDim.x + threadIdx.x;
    if (t >= nthreads) return;
    int e  = t >> 4;
    int f4 = (t & 15) << 2;
    float w = norm[e];
    int s = src[e], d = dst[e];
    float4 mv = *(const float4*)(M + (size_t)s * H_DIM + f4);
    float* ap = Agg + (size_t)d * H_DIM + f4;
    atomicAdd(ap + 0, mv.x * w);
    atomicAdd(ap + 1, mv.y * w);
    atomicAdd(ap + 2, mv.z * w);
    atomicAdd(ap + 3, mv.w * w);
}

// ------------------------------------------------------------------- pooling
// ReLU fused on load; values >= 0 so u32 bit-pattern atomicMax == float max.
__global__ void pool_max_kernel(const float* __restrict__ h,
                                const int* __restrict__ batch,
                                unsigned int* __restrict__ pooled_u, int n) {
    int t = blockIdx.x * blockDim.x + threadIdx.x;
    if (t >= n) return;
    int i = t >> 6;       // node
    int f = t & 63;       // feature
    float v = fmaxf(h[t], 0.0f);
    atomicMax(&pooled_u[(size_t)batch[i] * H_DIM + f], __float_as_uint(v));
}

// --------------------------------------------------------------- final linear
__global__ void final_linear_kernel(const float* __restrict__ pooled,
                                    const float* __restrict__ Wlin,
                                    const float* __restrict__ blin,
                                    float* __restrict__ out, int n) {
    int t = blockIdx.x * blockDim.x + threadIdx.x;
    if (t >= n) return;
    int g = t / C_OUT, c = t % C_OUT;
    float acc = blin[c];
    const float* pr = pooled + (size_t)g * H_DIM;
#pragma unroll
    for (int f = 0; f < H_DIM; ++f)
        acc = fmaf(pr[f], Wlin[f * C_OUT + c], acc);
    out[t] = acc;
}

// ==========================================================================
extern "C" void kernel_launch(void* const* d_in, const int* in_sizes, int n_in,
                              void* d_out, int out_size, void* d_ws, size_t ws_size,
                              hipStream_t stream) {
    (void)in_sizes; (void)n_in; (void)out_size; (void)ws_size;

    const float* x    = (const float*)d_in[0];
    const float* Wl[5] = { (const float*)d_in[1], (const float*)d_in[3],
                           (const float*)d_in[5], (const float*)d_in[7],
                           (const float*)d_in[9] };
    const float* bl[5] = { (const float*)d_in[2], (const float*)d_in[4],
                           (const float*)d_in[6], (const float*)d_in[8],
                           (const float*)d_in[10] };
    const float* Wlin = (const float*)d_in[11];
    const float* blin = (const float*)d_in[12];
    const int* edge_index = (const int*)d_in[13];
    const int* batch      = (const int*)d_in[14];
    const int* src = edge_index;             // edge_index[0, :]
    const int* dst = edge_index + N_EDGES;   // edge_index[1, :]
    float* out = (float*)d_out;

    // workspace layout (floats):
    //   ds | dinv | norm | P | Q | pooled | xpad | W1pad   (~65 MB)
    float* ws     = (float*)d_ws;
    float* ds     = ws;                                   // N   (deg -> dinv_sqrt)
    float* dinv   = ds + N_NODES;                         // N
    float* norm   = dinv + N_NODES;                       // E
    float* P      = norm + N_EDGES;                       // N*H (agg / h)
    float* Q      = P + (size_t)N_NODES * H_DIM;          // N*H (m = hW)
    float* pooled = Q + (size_t)N_NODES * H_DIM;          // G*H
    float* xpad   = pooled + (size_t)G_GRAPHS * H_DIM;    // N*16
    float* W1pad  = xpad + (size_t)N_NODES * K_PAD;       // 16*64

    const int TB = 256;
    // degrees + per-edge norm
    fill_kernel<<<(N_NODES + TB - 1) / TB, TB, 0, stream>>>(ds, 1.0f, N_NODES);
    deg_scatter_kernel<<<(N_EDGES + TB - 1) / TB, TB, 0, stream>>>(dst, ds, N_EDGES);
    deg_finish_kernel<<<(N_NODES + TB - 1) / TB, TB, 0, stream>>>(ds, dinv, N_NODES);
    edge_norm_kernel<<<(N_EDGES + TB - 1) / TB, TB, 0, stream>>>(src, dst, ds, norm, N_EDGES);

    // zero-padded layer-1 operands (removes all predication from the GEMM)
    pad_x_kernel<<<(N_NODES * K_PAD + TB - 1) / TB, TB, 0, stream>>>(
        x, xpad, N_NODES * K_PAD);
    pad_w_kernel<<<(K_PAD * H_DIM + TB - 1) / TB, TB, 0, stream>>>(Wl[0], W1pad);

    const int rowTiles = N_NODES / 16;            // 6250 (exact)
    const int scatterN = N_EDGES * 16;            // 16 threads / edge
    const int poolN    = N_NODES * H_DIM;

    for (int l = 0; l < 5; ++l) {
        // GEMM with fused self-loop/bias epilogue: Q = m, P = m*dinv + b
        if (l == 0)
            gemm_wmma_fused_kernel<K_PAD, false><<<rowTiles, 128, 0, stream>>>(
                xpad, W1pad, bl[0], dinv, Q, P);
        else
            gemm_wmma_fused_kernel<H_DIM, true><<<rowTiles, 128, 0, stream>>>(
                P, Wl[l], bl[l], dinv, Q, P);
        edge_scatter_kernel<<<(scatterN + TB - 1) / TB, TB, 0, stream>>>(
            Q, src, dst, norm, P, scatterN);
    }

    // segment-max pooling (ReLU fused) + classifier head
    fill_kernel<<<(G_GRAPHS * H_DIM + TB - 1) / TB, TB, 0, stream>>>(pooled, 0.0f,
                                                                     G_GRAPHS * H_DIM);
    pool_max_kernel<<<(poolN + TB - 1) / TB, TB, 0, stream>>>(
        P, batch, (unsigned int*)pooled, poolN);
    final_linear_kernel<<<(G_GRAPHS * C_OUT + TB - 1) / TB, TB, 0, stream>>>(
        pooled, Wlin, blin, out, G_GRAPHS * C_OUT);
}